// PostProcess_50826642981473
// MI455X (gfx1250) — compile-verified
//
#include <hip/hip_runtime.h>

// ---------------------------------------------------------------------------
// YOLO-style post-process (conf filter -> top-2048 -> class-offset NMS -> 300)
// for MI455X / gfx1250.
//  * Streaming stage: double-buffered Tensor Data Mover pipeline
//    (tensor_load_to_lds, 2 ops in flight, s_wait_tensorcnt 1/0).
//  * Box conversion cxcywh->xyxy done as a 16x4 @ 4x16 matrix product with
//    V_WMMA_F32_16X16X4_F32 (bit-identical to the scalar form: each output is
//    one product + one exact add of zeros under RNE).
//  * NMS: parallel suppression bit-matrix + wave32-resident 2048-bit greedy
//    scan with global_prefetch to hide the dependent row-load latency.
// ---------------------------------------------------------------------------

#define NA     25200
#define NCH    85
#define NC     80
#define CONF   0.96f
#define IOU_T  0.45f
#define CAP    4096
#define TOPK   2048
#define DETS   300

#define TILE_A        64                      // anchors per TDM tile (21760 B)
#define NTILES        ((NA + TILE_A - 1) / TILE_A)   // 394
#define FILTER_BLOCKS 64                      // blocks per image (grid-stride tiles)

struct Cand { float s; int idx; };

typedef unsigned int u32x4 __attribute__((ext_vector_type(4)));
typedef int          i32x4 __attribute__((ext_vector_type(4)));
typedef int          i32x8 __attribute__((ext_vector_type(8)));
typedef float        v2f   __attribute__((ext_vector_type(2)));
typedef float        v8f   __attribute__((ext_vector_type(8)));

#if defined(__has_builtin)
#if __has_builtin(__builtin_amdgcn_tensor_load_to_lds) && \
    __has_builtin(__builtin_amdgcn_s_wait_tensorcnt)
#define HAVE_TDM 1
#endif
#if __has_builtin(__builtin_amdgcn_wmma_f32_16x16x4_f32)
#define HAVE_WMMA 1
#endif
#endif

typedef __attribute__((address_space(3))) char lds_char_t;

__device__ __forceinline__ unsigned lds_byte_offset(const void* p) {
  // AS3 pointers are 32-bit LDS byte offsets within the workgroup allocation.
  return (unsigned)(unsigned long long)(lds_char_t*)(p);
}

#ifdef HAVE_TDM
// 1-D DMA of n fp32 elements from global memory into LDS via the TDM.
// D# layout per CDNA5 ISA ch.8: group0 = {count=1, lds_addr, global_addr, type=2},
// group1 = {data_size=4B, tensor_dim0=n, tile_dim0=n, tile_dim1=1, stride=n}.
__device__ __forceinline__ void tdm_load_1d_f32(unsigned lds_off, const float* src,
                                                unsigned n) {
  unsigned long long ga = (unsigned long long)(const void*)src;
  u32x4 g0;
  g0[0] = 1u;                                           // count=1, user descriptor
  g0[1] = lds_off;                                      // lds_addr
  g0[2] = (unsigned)(ga & 0xFFFFFFFFull);               // global_addr[31:0]
  g0[3] = (unsigned)((ga >> 32) & 0x1FFFFFFull)         // global_addr[56:32]
          | (2u << 30);                                 // type = 2 ("image")
  i32x8 g1;
  g1[0] = (int)(2u << 16);                              // data_size = 2 (4 bytes)
  g1[1] = (int)((n & 0xFFFFu) << 16);                   // tensor_dim0 lo16
  g1[2] = (int)(((n >> 16) & 0xFFFFu) | (1u << 16));    // tensor_dim0 hi16 | tensor_dim1=1
  g1[3] = (int)((n & 0xFFFFu) << 16);                   // tile_dim0 = n (n <= 65535)
  g1[4] = 1;                                            // tile_dim1 = 1
  g1[5] = (int)n;                                       // tensor_dim0_stride lo32
  g1[6] = 0;
  g1[7] = 0;
  i32x4 g2; g2[0] = 1; g2[1] = 1; g2[2] = 0; g2[3] = 0; // tensor_dim2/3 = 1
  i32x4 g3; g3[0] = 0; g3[1] = 0; g3[2] = 0; g3[3] = 0;
#if defined(__clang_major__) && (__clang_major__ >= 23)
  i32x8 g4; g4[0]=g4[1]=g4[2]=g4[3]=g4[4]=g4[5]=g4[6]=g4[7]=0;
  __builtin_amdgcn_tensor_load_to_lds(g0, g1, g2, g3, g4, 0);
#else
  __builtin_amdgcn_tensor_load_to_lds(g0, g1, g2, g3, 0);
#endif
}
#endif

// ---------------------------------------------------------------------------
__global__ void init_kernel(int* __restrict__ counters, int B) {
  const int t = blockIdx.x * blockDim.x + threadIdx.x;
  if (t < B) counters[t] = 0;
}

// Stage 1: double-buffered TDM pipeline staging prediction tiles into LDS,
// emitting candidates (score, flat index). 128 threads = 2 threads/anchor.
__global__ void filter_kernel(const float* __restrict__ pred,
                              Cand* __restrict__ cands,
                              int* __restrict__ counters) {
  __shared__ float tile[2][TILE_A * NCH];
  const int img  = blockIdx.y;
  const int t0   = blockIdx.x;
  const float* base = pred + (long long)img * NA * NCH;

#ifdef HAVE_TDM
  // ---- prologue: issue first tile into buffer 0 ----
  if (threadIdx.x < 32) {
    const int a0 = t0 * TILE_A;
    const unsigned n = (unsigned)(min(TILE_A, NA - a0) * NCH);
    tdm_load_1d_f32(lds_byte_offset(tile[0]), base + (long long)a0 * NCH, n);
  }
  int buf = 0;
  for (int t = t0; t < NTILES; t += FILTER_BLOCKS) {
    const int tn = t + FILTER_BLOCKS;
    if (threadIdx.x < 32) {
      if (tn < NTILES) {
        const int an = tn * TILE_A;
        const unsigned nn = (unsigned)(min(TILE_A, NA - an) * NCH);
        tdm_load_1d_f32(lds_byte_offset(tile[buf ^ 1]), base + (long long)an * NCH, nn);
        __builtin_amdgcn_s_wait_tensorcnt(1);   // current tile complete
      } else {
        __builtin_amdgcn_s_wait_tensorcnt(0);   // last tile complete
      }
    }
    __syncthreads();                            // tile[buf] visible to all waves

    const int a0    = t * TILE_A;
    const int tileA = min(TILE_A, NA - a0);
    const int a     = threadIdx.x >> 1;
    if (a < tileA) {
      const float* row = &tile[buf][a * NCH];
      const float obj = row[4];
      if (obj > CONF) {
        const int cbase = (threadIdx.x & 1) * (NC / 2);
        for (int c = 0; c < NC / 2; ++c) {
          const float s = row[5 + cbase + c] * obj;
          if (s > CONF) {
            const int pos = atomicAdd(&counters[img], 1);
            if (pos < CAP) {
              Cand cd; cd.s = s; cd.idx = (a0 + a) * NC + (cbase + c);
              cands[img * CAP + pos] = cd;
            }
          }
        }
      }
    }
    __syncthreads();                            // done reading before buf reuse
    buf ^= 1;
  }
#else
  for (int t = t0; t < NTILES; t += FILTER_BLOCKS) {
    const int a0    = t * TILE_A;
    const int tileA = min(TILE_A, NA - a0);
    const unsigned n = (unsigned)(tileA * NCH);
    const float* gsrc = base + (long long)a0 * NCH;
    for (unsigned k = threadIdx.x; k < n; k += blockDim.x) tile[0][k] = gsrc[k];
    __syncthreads();
    const int a = threadIdx.x >> 1;
    if (a < tileA) {
      const float* row = &tile[0][a * NCH];
      const float obj = row[4];
      if (obj > CONF) {
        const int cbase = (threadIdx.x & 1) * (NC / 2);
        for (int c = 0; c < NC / 2; ++c) {
          const float s = row[5 + cbase + c] * obj;
          if (s > CONF) {
            const int pos = atomicAdd(&counters[img], 1);
            if (pos < CAP) {
              Cand cd; cd.s = s; cd.idx = (a0 + a) * NC + (cbase + c);
              cands[img * CAP + pos] = cd;
            }
          }
        }
      }
    }
    __syncthreads();
  }
#endif
}

// Stage 2: per-image bitonic sort of candidates by 64-bit key
// (ordered-score desc, index asc) -> top-2048 scores/labels/boxes.
// Box conversion uses V_WMMA_F32_16X16X4_F32: D(16x16) = A(16x4) x B(4x16),
// A rows = candidate {cx,cy,w,h}, B cols 0..3 = the xyxy linear map.
__global__ void sort_kernel(const float* __restrict__ pred,
                            const Cand* __restrict__ cands,
                            const int* __restrict__ counters,
                            float* __restrict__ scores,
                            int* __restrict__ labels,
                            float4* __restrict__ boxes) {
  __shared__ unsigned long long key[CAP];
#ifdef HAVE_WMMA
  __shared__ float stage[32][16][4];            // per-wave D unscramble buffer
#endif
  const int img = blockIdx.x;
  const int n = min(counters[img], CAP);

  for (int k = threadIdx.x; k < CAP; k += blockDim.x) {
    unsigned long long kk = 0ull;
    if (k < n) {
      const Cand c = cands[img * CAP + k];
      unsigned sb = __float_as_uint(c.s);
      sb = (sb & 0x80000000u) ? ~sb : (sb | 0x80000000u);   // total order on floats
      kk = ((unsigned long long)sb << 32)
         | (unsigned long long)(0xFFFFFFFFu - (unsigned)c.idx);
    }
    key[k] = kk;
  }
  __syncthreads();

  for (int ksz = 2; ksz <= CAP; ksz <<= 1) {
    for (int j = ksz >> 1; j > 0; j >>= 1) {
      for (int i = threadIdx.x; i < CAP; i += blockDim.x) {
        const int ixj = i ^ j;
        if (ixj > i) {
          const bool descend = ((i & ksz) == 0);
          const unsigned long long a = key[i], b = key[ixj];
          if (descend ? (a < b) : (a > b)) { key[i] = b; key[ixj] = a; }
        }
      }
      __syncthreads();
    }
  }

#ifdef HAVE_WMMA
  // 32 waves x 4 groups of 16 candidates = 2048. Full-wave EXEC for WMMA.
  const int wv   = threadIdx.x >> 5;
  const int lane = threadIdx.x & 31;
  const int nn   = lane & 15;
  // Constant B (4x16): lanes 0..15 hold rows K0,K1; lanes 16..31 rows K2,K3.
  v2f bmat;
  if (lane < 16) {
    bmat.x = (nn == 0 || nn == 2) ? 1.0f : 0.0f;                 // K0 = cx
    bmat.y = (nn == 1 || nn == 3) ? 1.0f : 0.0f;                 // K1 = cy
  } else {
    bmat.x = (nn == 0) ? -0.5f : (nn == 2) ? 0.5f : 0.0f;        // K2 = w
    bmat.y = (nn == 1) ? -0.5f : (nn == 3) ? 0.5f : 0.0f;        // K3 = h
  }
  for (int gg = wv; gg < TOPK / 16; gg += 32) {
    const int k = gg * 16 + nn;
    const unsigned long long kk = key[k];
    const bool valid = (kk != 0ull);
    int anchor = 0, label = 0; float score = -1.0f;
    if (valid) {
      const unsigned sb = (unsigned)(kk >> 32);
      const unsigned fb = (sb & 0x80000000u) ? (sb & 0x7FFFFFFFu) : ~sb;
      score = __uint_as_float(fb);
      const int idx = (int)(0xFFFFFFFFu - (unsigned)(kk & 0xFFFFFFFFu));
      anchor = idx / NC;
      label = idx - anchor * NC;
    }
    const float* p = pred + ((long long)img * NA + anchor) * NCH;
    v2f amat;                                   // A row = [cx cy w h]
    if (lane < 16) { amat.x = valid ? p[0] : 0.0f; amat.y = valid ? p[1] : 0.0f; }
    else           { amat.x = valid ? p[2] : 0.0f; amat.y = valid ? p[3] : 0.0f; }
    v8f d = {0.f, 0.f, 0.f, 0.f, 0.f, 0.f, 0.f, 0.f};
    d = __builtin_amdgcn_wmma_f32_16x16x4_f32(false, amat, false, bmat,
                                              (short)0, d, false, false);
    // D layout: VGPR g -> rows g (lanes 0..15) / g+8 (lanes 16..31), col = lane&15.
    if (nn < 4) {
      const int rbase = (lane < 16) ? 0 : 8;
      #pragma unroll
      for (int g = 0; g < 8; ++g) stage[wv][rbase + g][nn] = d[g];
    }
    __builtin_amdgcn_wave_barrier();            // keep LDS write->read order
    if (lane < 16) {
      const float4 bx = valid
          ? make_float4(stage[wv][nn][0], stage[wv][nn][1],
                        stage[wv][nn][2], stage[wv][nn][3])
          : make_float4(0.f, 0.f, 0.f, 0.f);
      scores[img * TOPK + k] = score;
      labels[img * TOPK + k] = label;
      boxes[img * TOPK + k]  = bx;
    }
    __builtin_amdgcn_wave_barrier();            // reads done before next overwrite
  }
#else
  for (int k = threadIdx.x; k < TOPK; k += blockDim.x) {
    const unsigned long long kk = key[k];
    float score; int label; float4 bx;
    if (kk == 0ull) {
      score = -1.0f; label = 0; bx = make_float4(0.f, 0.f, 0.f, 0.f);
    } else {
      const unsigned sb = (unsigned)(kk >> 32);
      const unsigned fb = (sb & 0x80000000u) ? (sb & 0x7FFFFFFFu) : ~sb;
      score = __uint_as_float(fb);
      const int idx = (int)(0xFFFFFFFFu - (unsigned)(kk & 0xFFFFFFFFu));
      const int anchor = idx / NC;
      label = idx - anchor * NC;
      const float* p = pred + ((long long)img * NA + anchor) * NCH;
      const float cx = p[0], cy = p[1], w = p[2], h = p[3];
      bx = make_float4(cx - 0.5f * w, cy - 0.5f * h, cx + 0.5f * w, cy + 0.5f * h);
    }
    scores[img * TOPK + k] = score;
    labels[img * TOPK + k] = label;
    boxes[img * TOPK + k]  = bx;
  }
#endif
}

// Stage 3: suppression bit-matrix. One wave32 per row i; lane l owns u64 word l
// (columns j = 64l..64l+63). Bit set iff j>i and IoU(offset boxes) > 0.45.
__global__ void mask_kernel(const int* __restrict__ labels,
                            const float4* __restrict__ boxes,
                            unsigned long long* __restrict__ maskmat) {
  const int i    = blockIdx.x;
  const int img  = blockIdx.y;
  const int lane = threadIdx.x;
  const float4* bx = boxes + img * TOPK;
  const int*    lb = labels + img * TOPK;

  const float4 bi  = bx[i];
  const float offi = 4.0f * (float)lb[i];
  const float ax1 = bi.x + offi, ay1 = bi.y + offi;
  const float ax2 = bi.z + offi, ay2 = bi.w + offi;
  const float area_a = (ax2 - ax1) * (ay2 - ay1);

  unsigned long long m = 0ull;
  const int jbase = lane * 64;
  if (jbase + 63 > i) {
    for (int b = 0; b < 64; ++b) {
      const int j = jbase + b;
      if (j <= i) continue;
      const float4 bj  = bx[j];
      const float offj = 4.0f * (float)lb[j];
      const float bx1 = bj.x + offj, by1 = bj.y + offj;
      const float bx2 = bj.z + offj, by2 = bj.w + offj;
      const float ltx = fmaxf(ax1, bx1), lty = fmaxf(ay1, by1);
      const float rbx = fminf(ax2, bx2), rby = fminf(ay2, by2);
      const float iw = fmaxf(rbx - ltx, 0.0f), ih = fmaxf(rby - lty, 0.0f);
      const float inter  = iw * ih;
      const float area_b = (bx2 - bx1) * (by2 - by1);
      const float iou = inter / (area_a + area_b - inter + 1e-9f);
      if (iou > IOU_T) m |= (1ull << b);
    }
  }
  maskmat[((long long)img * TOPK + i) * 32 + lane] = m;
}

// Stage 4: sequential greedy scan with the 2048-bit keep mask held across one
// wave32 (lane l = word l), then wave prefix-popcount compaction to 300 rows.
__global__ void nms_scan_out_kernel(const float* __restrict__ scores,
                                    const int* __restrict__ labels,
                                    const float4* __restrict__ boxes,
                                    const unsigned long long* __restrict__ maskmat,
                                    float* __restrict__ out) {
  const int img  = blockIdx.x;
  const int lane = threadIdx.x;
  const float* sc = scores + img * TOPK;

  unsigned long long remv = 0ull;                // bit set => removed/invalid
  for (int b = 0; b < 64; ++b) {
    const int j = lane * 64 + b;
    if (!(sc[j] > 0.0f)) remv |= (1ull << b);
  }

  const unsigned long long* mm = maskmat + (long long)img * TOPK * 32;
  for (int i = 0; i < TOPK; ++i) {
    if (i + 1 < TOPK) {
      // hide the dependent-load latency of the next mask row
      __builtin_prefetch((const void*)&mm[(long long)(i + 1) * 32 + lane], 0, 0);
    }
    const unsigned long long w = __shfl(remv, i >> 6, 32);   // uniform broadcast
    if (((w >> (i & 63)) & 1ull) == 0ull) {                  // i is kept
      remv |= mm[(long long)i * 32 + lane];
    }
  }
  const unsigned long long keep = ~remv;

  const int cnt = __popcll(keep);
  int incl = cnt;
  for (int d = 1; d < 32; d <<= 1) {
    const int v = __shfl_up(incl, d, 32);
    if (lane >= d) incl += v;
  }
  const int excl  = incl - cnt;
  const int total = __shfl(incl, 31, 32);

  float* o = out + (long long)img * DETS * 6;
  int r = excl;
  for (int b = 0; b < 64; ++b) {
    if ((keep >> b) & 1ull) {
      if (r < DETS) {
        const int j = lane * 64 + b;
        const float4 bx = boxes[img * TOPK + j];
        o[r * 6 + 0] = bx.x; o[r * 6 + 1] = bx.y;
        o[r * 6 + 2] = bx.z; o[r * 6 + 3] = bx.w;
        o[r * 6 + 4] = sc[j];
        o[r * 6 + 5] = (float)labels[img * TOPK + j];
      }
      ++r;
    }
  }
  for (int rr = total + lane; rr < DETS; rr += 32) {
    #pragma unroll
    for (int q = 0; q < 6; ++q) o[rr * 6 + q] = 0.0f;
  }
}

// ---------------------------------------------------------------------------
extern "C" void kernel_launch(void* const* d_in, const int* in_sizes, int n_in,
                              void* d_out, int out_size, void* d_ws, size_t ws_size,
                              hipStream_t stream) {
  (void)n_in; (void)out_size; (void)ws_size;
  const float* pred = (const float*)d_in[0];
  const int B = in_sizes[0] / (NA * NCH);       // 8 for the reference shapes

  char* ws = (char*)d_ws;
  size_t off = 0;
  int*   counters = (int*)(ws + off);                 off += 256;
  Cand*  cands    = (Cand*)(ws + off);                off += (size_t)B * CAP * sizeof(Cand);
  float* scores   = (float*)(ws + off);               off += (size_t)B * TOPK * sizeof(float);
  int*   labels   = (int*)(ws + off);                 off += (size_t)B * TOPK * sizeof(int);
  float4* boxes   = (float4*)(ws + off);              off += (size_t)B * TOPK * sizeof(float4);
  unsigned long long* maskmat = (unsigned long long*)(ws + off);
  off += (size_t)B * TOPK * 32 * sizeof(unsigned long long);

  init_kernel<<<1, 256, 0, stream>>>(counters, B);
  filter_kernel<<<dim3(FILTER_BLOCKS, B), 128, 0, stream>>>(pred, cands, counters);
  sort_kernel<<<B, 1024, 0, stream>>>(pred, cands, counters, scores, labels, boxes);
  mask_kernel<<<dim3(TOPK, B), 32, 0, stream>>>(labels, boxes, maskmat);
  nms_scan_out_kernel<<<B, 32, 0, stream>>>(scores, labels, boxes, maskmat, (float*)d_out);
}